// LorentzBatchNorm_DistVar_21199958573192
// MI455X (gfx1250) — compile-verified
//
#include <hip/hip_runtime.h>
#include <hip/hip_bf16.h>
#include <stdint.h>

// ---------------- problem dims (from reference setup_inputs) ----------------
#define B_DIM   64
#define T_DIM   1024
#define D_DIM   256
#define NPTS    (B_DIM * T_DIM)        // 65536 points on the hyperboloid
#define TSPLIT  8
#define TCHUNK  (T_DIM / TSPLIT)       // 128 tokens per partial block

// ---------------- workspace layout (float indices) ----------------
#define WS_VARACC 0                    // scalar: sum of clipped squared distances
#define WS_C0     1                    // scalar: 1 / (1 - <mean, beta>_L)
#define WS_MEAN   16                   // mean[256]
#define WS_MEANN  288                  // mean with time component negated [256]
#define WS_VPART  2048                 // per-block variance partials [1024]
#define WS_PART   4096                 // token partial sums [B][TSPLIT][D] = 131072

typedef __attribute__((ext_vector_type(2))) float v2f;
typedef __attribute__((ext_vector_type(8))) float v8f;

// 128-bit payload type expected by the async global->LDS builtin
typedef __attribute__((vector_size(4 * sizeof(int)))) int v4i_t;
typedef __attribute__((address_space(1))) v4i_t gv4i_t;   // global
typedef __attribute__((address_space(3))) v4i_t lv4i_t;   // LDS

// ---------------- helpers ----------------
__device__ __forceinline__ float waveSum(float v) {
#pragma unroll
  for (int off = 16; off > 0; off >>= 1) v += __shfl_xor(v, off, 32);
  return v;
}

__device__ __forceinline__ float eucDot8(float4 a0, float4 a1, float4 b0, float4 b1) {
  return a0.x*b0.x + a0.y*b0.y + a0.z*b0.z + a0.w*b0.w +
         a1.x*b1.x + a1.y*b1.y + a1.z*b1.z + a1.w*b1.w;
}
// Minkowski partial: lane 0 owns dims 0..7 which include the time dim (d=0)
__device__ __forceinline__ float minkDot8(float4 a0, float4 a1, float4 b0, float4 b1, int lane) {
  float s = eucDot8(a0, a1, b0, b1);
  if (lane == 0) s -= 2.0f * a0.x * b0.x;
  return s;
}
__device__ __forceinline__ float4 f4axpy(float4 a, float s, float4 b) {  // a + s*b
  return make_float4(fmaf(s, b.x, a.x), fmaf(s, b.y, a.y), fmaf(s, b.z, a.z), fmaf(s, b.w, a.w));
}
__device__ __forceinline__ float4 f4scale(float4 a, float s) {
  return make_float4(a.x * s, a.y * s, a.z * s, a.w * s);
}
__device__ __forceinline__ float4 f4add(float4 a, float4 b) {
  return make_float4(a.x + b.x, a.y + b.y, a.z + b.z, a.w + b.w);
}

// ---------------- K1: partial token sums per (batch, T-slice) ----------------
__global__ __launch_bounds__(256) void k1_partial(const float* __restrict__ x,
                                                  float* __restrict__ ws) {
  const int d = threadIdx.x;                 // 0..255 (one dim per thread, coalesced)
  const int b = blockIdx.x;                  // 0..63
  const int s = blockIdx.y;                  // 0..TSPLIT-1
  const float* p = x + ((size_t)b * T_DIM + (size_t)s * TCHUNK) * D_DIM + d;
  float acc = 0.0f;
#pragma unroll 4
  for (int t = 0; t < TCHUNK; ++t) acc += p[(size_t)t * D_DIM];
  ws[WS_PART + (b * TSPLIT + s) * D_DIM + d] = acc;
}

// ---------------- K2: double centroid -> mean, plus constants ----------------
__global__ __launch_bounds__(256) void k2_mean(const float* __restrict__ beta,
                                               float* __restrict__ ws) {
  __shared__ float red[256];
  const int d = threadIdx.x;
  float acc = 0.0f;
  for (int b = 0; b < B_DIM; ++b) {
    float m = 0.0f;
#pragma unroll
    for (int s = 0; s < TSPLIT; ++s) m += ws[WS_PART + (b * TSPLIT + s) * D_DIM + d];
    m *= (1.0f / T_DIM);                       // token mean
    float q = m * m; if (d == 0) q = -q;       // <m,m>_L partial
    red[d] = q; __syncthreads();
    for (int st = 128; st > 0; st >>= 1) { if (d < st) red[d] += red[d + st]; __syncthreads(); }
    float li = red[0]; __syncthreads();
    acc += m / sqrtf(fmaxf(-li, 1e-8f));       // normalized centroid, accumulate over B
  }
  float mv = acc * (1.0f / B_DIM);
  float q = mv * mv; if (d == 0) q = -q;
  red[d] = q; __syncthreads();
  for (int st = 128; st > 0; st >>= 1) { if (d < st) red[d] += red[d + st]; __syncthreads(); }
  float li = red[0]; __syncthreads();
  const float mean = mv / sqrtf(fmaxf(-li, 1e-8f));
  ws[WS_MEAN + d]  = mean;
  ws[WS_MEANN + d] = (d == 0) ? -mean : mean;
  // c0 = 1 / (1 - <mean, beta>_L)
  float bb = beta[d];
  q = mean * bb; if (d == 0) q = -q;
  red[d] = q; __syncthreads();
  for (int st = 128; st > 0; st >>= 1) { if (d < st) red[d] += red[d + st]; __syncthreads(); }
  if (d == 0) ws[WS_C0] = 1.0f / (1.0f - red[0]);
}

// ---------------- K3: variance partials via WMMA f32 16x16x4 ----------------
// 64-point tile staged into LDS with async global->LDS b128 copies; each of the
// 4 waves computes <x_p, mean>_L for its 16 points as a 16x16x(K=256) WMMA
// matvec (B columns all broadcast the time-negated mean chunk).
#define ROWS    64
#define RSTRIDE 260   // 256 + 4-float pad -> conflict-free strided ds reads
__global__ __launch_bounds__(128) void k3_var(const float* __restrict__ x,
                                              float* __restrict__ ws) {
  __shared__ float tile[ROWS * RSTRIDE];
  __shared__ float meanS[D_DIM];
  __shared__ float bacc[4];
  const int tid  = threadIdx.x;
  const int lane = tid & 31;
  const int wv   = tid >> 5;

  meanS[tid]       = ws[WS_MEANN + tid];
  meanS[tid + 128] = ws[WS_MEANN + tid + 128];

  // stage 64 KiB tile: 4096 float4s, 128 threads -> 32 rounds, fully coalesced
  const size_t base = (size_t)blockIdx.x * (ROWS * D_DIM);
#pragma unroll
  for (int j = 0; j < 32; ++j) {
    const int idx4 = j * 128 + tid;
    const int e    = idx4 * 4;
    const int row  = e >> 8;
    const int col  = e & 255;
    const float* gp = x + base + e;
    float*       lp = &tile[row * RSTRIDE + col];
#if __has_builtin(__builtin_amdgcn_global_load_async_to_lds_b128)
    __builtin_amdgcn_global_load_async_to_lds_b128((gv4i_t*)(uintptr_t)gp,
                                                   (lv4i_t*)(uintptr_t)lp, 0, 0);
#else
    asm volatile("global_load_async_to_lds_b128 %0, %1, off"
                 :: "v"((unsigned)(uintptr_t)lp), "v"(gp) : "memory");
#endif
  }
#if __has_builtin(__builtin_amdgcn_s_wait_asynccnt)
  __builtin_amdgcn_s_wait_asynccnt(0);
#else
  asm volatile("s_wait_asynccnt 0" ::: "memory");
#endif
  __syncthreads();

  // WMMA A layout (16x4 f32): lanes 0-15 hold K=0,1; lanes 16-31 hold K=2,3; M = lane%16
  const int halfk   = lane >> 4;
  const int rowbase = (wv * 16 + (lane & 15)) * RSTRIDE;
  v8f c = {0.f, 0.f, 0.f, 0.f, 0.f, 0.f, 0.f, 0.f};
#pragma unroll 4
  for (int k = 0; k < 64; ++k) {
    const int col = 4 * k + 2 * halfk;
    v2f a;  a.x  = tile[rowbase + col];  a.y  = tile[rowbase + col + 1];
    v2f bb; bb.x = meanS[col];           bb.y = meanS[col + 1];   // broadcast columns
    c = __builtin_amdgcn_wmma_f32_16x16x4_f32(false, a, false, bb, (short)0, c,
                                              false, false);
  }
  // D layout: lanes 0-15 VGPR r = point r; lanes 16-31 VGPR r = point 8+r
  float part = 0.0f;
#pragma unroll
  for (int r = 0; r < 8; ++r) {
    const float xy   = c[r];                         // <x_p, mean>_L
    const float aarg = fmaxf(-xy, 1.0f + 1e-7f);
    const float dd   = acoshf(aarg);
    part += fmaxf(dd * dd, 1e-8f);
  }
  float contrib = ((lane & 15) == 0) ? part : 0.0f;  // lane0 + lane16 = 16 points
  contrib = waveSum(contrib);
  if (lane == 0) bacc[wv] = contrib;
  __syncthreads();
  if (tid == 0) ws[WS_VPART + blockIdx.x] = bacc[0] + bacc[1] + bacc[2] + bacc[3];
}

// ---------------- K3b: deterministic reduce of variance partials ----------------
__global__ __launch_bounds__(256) void k3b_reduce(float* __restrict__ ws) {
  __shared__ float red[256];
  const int t = threadIdx.x;
  float s = 0.0f;
  for (int i = t; i < NPTS / ROWS; i += 256) s += ws[WS_VPART + i];
  red[t] = s; __syncthreads();
  for (int st = 128; st > 0; st >>= 1) { if (t < st) red[t] += red[t + st]; __syncthreads(); }
  if (t == 0) ws[WS_VARACC] = red[0];
}

// ---------------- K4: logmap/scale/cap/transport/expmap, 1 point per wave ----------------
__global__ __launch_bounds__(256) void k4_out(const float* __restrict__ x,
                                              const float* __restrict__ beta,
                                              const float* __restrict__ gamma,
                                              const float* __restrict__ ws,
                                              float* __restrict__ out) {
  const int lane = threadIdx.x & 31;
  const int wv   = threadIdx.x >> 5;
  const size_t p = (size_t)blockIdx.x * 8 + wv;     // point index, exact cover
  const size_t g = p * D_DIM + (size_t)lane * 8;    // wave reads a contiguous 1 KiB row

  const float4 x0 = *(const float4*)(x + g);
  const float4 x1 = *(const float4*)(x + g + 4);
  const float4 m0 = *(const float4*)(ws + WS_MEAN + lane * 8);
  const float4 m1 = *(const float4*)(ws + WS_MEAN + lane * 8 + 4);
  const float4 b0 = *(const float4*)(beta + lane * 8);
  const float4 b1 = *(const float4*)(beta + lane * 8 + 4);
  const float c0  = ws[WS_C0];
  const float var = sqrtf(ws[WS_VARACC] * (1.0f / NPTS));
  const float scale = gamma[0] / (var + 1e-5f);

  // logmap(mean, x)
  const float xy   = waveSum(minkDot8(m0, m1, x0, x1, lane));   // <mean,x>_L
  const float dd   = acoshf(fmaxf(-xy, 1.0f + 1e-7f));
  const float4 u0  = f4axpy(x0, xy, m0);                        // x + xy*mean
  const float4 u1  = f4axpy(x1, xy, m1);
  const float  uu  = waveSum(minkDot8(u0, u1, u0, u1, lane));
  const float  un  = sqrtf(fmaxf(uu, 1e-8f));
  const float  s1  = dd / un * scale;                           // fold in gamma/(var+eps)
  float4 t0 = f4scale(u0, s1), t1 = f4scale(u1, s1);

  // rescale_to_max_euclid (Euclidean norm over all D components)
  const float n = sqrtf(waveSum(eucDot8(t0, t1, t0, t1)));
  const float f = fminf(1.0f, 32.0f / fmaxf(n, 1e-8f));
  t0 = f4scale(t0, f); t1 = f4scale(t1, f);

  // parallel transport mean -> beta
  const float lv = waveSum(minkDot8(b0, b1, t0, t1, lane));     // <beta, v>_L
  const float cc = lv * c0;
  const float4 v0 = f4axpy(t0, cc, f4add(m0, b0));
  const float4 v1 = f4axpy(t1, cc, f4add(m1, b1));

  // expmap at beta
  const float vn = sqrtf(fmaxf(waveSum(minkDot8(v0, v1, v0, v1, lane)), 1e-8f));
  const float ch = coshf(vn), sh = sinhf(vn) / vn;
  *(float4*)(out + g)     = f4add(f4scale(b0, ch), f4scale(v0, sh));
  *(float4*)(out + g + 4) = f4add(f4scale(b1, ch), f4scale(v1, sh));
}

// ---------------- launch ----------------
extern "C" void kernel_launch(void* const* d_in, const int* in_sizes, int n_in,
                              void* d_out, int out_size, void* d_ws, size_t ws_size,
                              hipStream_t stream) {
  (void)in_sizes; (void)n_in; (void)out_size; (void)ws_size;
  const float* x     = (const float*)d_in[0];
  const float* beta  = (const float*)d_in[1];
  const float* gamma = (const float*)d_in[2];
  float* ws  = (float*)d_ws;
  float* out = (float*)d_out;

  hipLaunchKernelGGL(k1_partial, dim3(B_DIM, TSPLIT), dim3(256), 0, stream, x, ws);
  hipLaunchKernelGGL(k2_mean,    dim3(1),             dim3(256), 0, stream, beta, ws);
  hipLaunchKernelGGL(k3_var,     dim3(NPTS / ROWS),   dim3(128), 0, stream, x, ws);
  hipLaunchKernelGGL(k3b_reduce, dim3(1),             dim3(256), 0, stream, ws);
  hipLaunchKernelGGL(k4_out,     dim3(NPTS / 8),      dim3(256), 0, stream, x, beta, gamma, ws, out);
}